// NeuralAutoregressiveFlow_7146825581245
// MI455X (gfx1250) — compile-verified
//
#include <hip/hip_runtime.h>
#include <hip/hip_bf16.h>
#include <math.h>

// ---- problem constants (from reference) ----
#define E        1024
#define MLPW     256
#define SZ_FIRST 90112   // (3*8 + 64) * E
#define SZ_MID   147456  // (2*8 + 2*64) * E
#define SZ_LAST  11264   // (3 + 8) * E

typedef __attribute__((ext_vector_type(2))) float v2f;
typedef __attribute__((ext_vector_type(8))) float v8f;

__device__ __forceinline__ float softplusf_(float v) {
    // stable log1p(exp(v))
    return fmaxf(v, 0.0f) + log1pf(expf(-fabsf(v)));
}
__device__ __forceinline__ float sigmoidf_(float v) {
    return 1.0f / (1.0f + expf(-v));
}

// ------------------------------------------------------------------
// init: xf = x
// ------------------------------------------------------------------
__global__ void k_init(const float* __restrict__ x, float* __restrict__ xf) {
    int i = blockIdx.x * blockDim.x + threadIdx.x;
    if (i < E) xf[i] = x[i];
}

// ------------------------------------------------------------------
// mask split: x1 = m?xf:0, x2 = m?0:xf   (m = parity(n)==parity(ii))
// ------------------------------------------------------------------
__global__ void k_mask_split(const float* __restrict__ xf,
                             float* __restrict__ x1, float* __restrict__ x2, int ii) {
    int n = blockIdx.x * blockDim.x + threadIdx.x;
    if (n < E) {
        bool m = ((n & 1) == (ii & 1));
        float v = xf[n];
        x1[n] = m ? v : 0.0f;
        x2[n] = m ? 0.0f : v;
    }
}

// ------------------------------------------------------------------
// MLP layer 0: 1024 -> 256 GEMV, split-K (8 slices) x 2 mlp instances
// (jjsel 0 -> mlp jj=0, jjsel 1 -> mlp jj=2). Deterministic: partial
// sums to separate slots hpart[(jjsel*8+ks)*256 + j].
// ------------------------------------------------------------------
__global__ void k_mlp0(const float* __restrict__ x1, const float* __restrict__ w0,
                       float* __restrict__ hpart, int ii) {
    int jjsel = blockIdx.x >> 3;
    int ks    = blockIdx.x & 7;
    int jj    = jjsel ? 2 : 0;
    const float* __restrict__ W = w0 + (size_t)(ii * 3 + jj) * E * MLPW;
    __shared__ float xs[128];
    int j = threadIdx.x;
    int k0 = ks * 128;
    if (j < 128) xs[j] = x1[k0 + j];
    __syncthreads();
    float acc = 0.0f;
    const float* __restrict__ Wp = W + (size_t)k0 * MLPW + j;
#pragma unroll 8
    for (int k = 0; k < 128; ++k) {
        acc = fmaf(xs[k], *Wp, acc);
        Wp += MLPW;
    }
    hpart[(jjsel * 8 + ks) * MLPW + j] = acc;
}

// ------------------------------------------------------------------
// MLP layer 1: reduce partials + bias, tanh, 256x256 GEMV, tanh -> h
// ------------------------------------------------------------------
__global__ void k_mlp1(const float* __restrict__ hpart, const float* __restrict__ b0,
                       const float* __restrict__ w1, const float* __restrict__ b1,
                       float* __restrict__ h, int ii) {
    int jjsel = blockIdx.x;           // 0 or 1
    int jj    = jjsel ? 2 : 0;
    __shared__ float t[MLPW];
    int j = threadIdx.x;
    float s = b0[(ii * 3 + jj) * MLPW + j];
#pragma unroll
    for (int ks = 0; ks < 8; ++ks)
        s += hpart[(jjsel * 8 + ks) * MLPW + j];
    t[j] = tanhf(s);
    __syncthreads();
    const float* __restrict__ W = w1 + (size_t)(ii * 3 + jj) * MLPW * MLPW + j;
    float acc = b1[(ii * 3 + jj) * MLPW + j];
#pragma unroll 8
    for (int k = 0; k < MLPW; ++k) {
        acc = fmaf(t[k], *W, acc);
        W += MLPW;
    }
    h[jjsel * MLPW + j] = tanhf(acc);
}

// ------------------------------------------------------------------
// Big GEMV via V_WMMA_F32_16X16X4_F32.
// out[col] = sum_k h[k] * W[k, col] + bias[col],  W row-major (256 x out_cols).
// A (16x4): all 16 rows hold the same h chunk -> D row 0 is the dot product.
//   A layout (ISA 7.12.2): VGPR0 = K (lane<16 ? 0 : 2), VGPR1 = K (lane<16 ? 1 : 3).
// B (4x16): VGPR0 = row K{0|2}, cols N=lane&15 ; VGPR1 = row K{1|3}.
// Each wave: 16 output cols, K=256 in 64 WMMA steps, 2 alternating
// accumulators to break the D->C dependency chain. WG = 8 waves = 128 cols.
// Row pointers are walked with a loop-invariant stride (no per-load shifts).
// ------------------------------------------------------------------
__global__ void k_biggemv(const float* __restrict__ h, const float* __restrict__ W,
                          const float* __restrict__ bias, float* __restrict__ outp,
                          int out_cols) {
    __shared__ float hs[MLPW];
    int tid = threadIdx.x;
    hs[tid] = h[tid];
    __syncthreads();

    int lane  = tid & 31;
    int wave  = tid >> 5;
    int n0    = (blockIdx.x * 8 + wave) * 16;
    int lhalf = lane >> 4;       // 0: K{0,1}, 1: K{2,3}
    int lcol  = lane & 15;

    // per-lane row pointers: rp0 -> row (k0 + 2*lhalf), rp1 -> row (k0 + 2*lhalf + 1)
    const float* __restrict__ rp0 = W + (size_t)(2 * lhalf) * out_cols + n0 + lcol;
    const float* __restrict__ rp1 = rp0 + out_cols;
    const size_t rstep = (size_t)out_cols * 4;   // advance 4 K-rows

    v8f c0 = {0.f, 0.f, 0.f, 0.f, 0.f, 0.f, 0.f, 0.f};
    v8f c1 = {0.f, 0.f, 0.f, 0.f, 0.f, 0.f, 0.f, 0.f};

#pragma unroll 4
    for (int k0 = 0; k0 < MLPW; k0 += 8) {
        int kb = k0 + 2 * lhalf;
        v2f a0, b0, a1, b1;
        a0[0] = hs[kb];
        a0[1] = hs[kb + 1];
        // non-temporal: weights are streamed once (510 MB >> 192 MB L2)
        b0[0] = __builtin_nontemporal_load(rp0);
        b0[1] = __builtin_nontemporal_load(rp1);
        rp0 += rstep; rp1 += rstep;
        a1[0] = hs[kb + 4];
        a1[1] = hs[kb + 5];
        b1[0] = __builtin_nontemporal_load(rp0);
        b1[1] = __builtin_nontemporal_load(rp1);
        rp0 += rstep; rp1 += rstep;
        c0 = __builtin_amdgcn_wmma_f32_16x16x4_f32(
                false, a0, false, b0, (short)0, c0, false, false);
        c1 = __builtin_amdgcn_wmma_f32_16x16x4_f32(
                false, a1, false, b1, (short)0, c1, false, false);
    }
    // D row M=0 lives in element 0 of the accumulators on lanes 0..15 (N = lane)
    if (lane < 16)
        outp[n0 + lane] = c0[0] + c1[0] + bias[n0 + lane];
}

// ------------------------------------------------------------------
// Elementwise DSF (3 blocks, d=8) + forward-mode JVP, masked update of xf,
// per-block log-jacobian sums written to distinct slots (deterministic).
// Param layout: params[p] is (rows, E); value (t, n) at p[t*E + n].
// ------------------------------------------------------------------
__global__ void k_dsf(const float* __restrict__ p0, const float* __restrict__ p1,
                      const float* __restrict__ p2, const float* __restrict__ x2,
                      float* __restrict__ xf, float* __restrict__ ljd_arr, int ii) {
    int n = blockIdx.x * blockDim.x + threadIdx.x;

    float h[8], dh[8];
    // ---- block 0: d0=1, d1=8 : rows a[0:8) b[8:16) w[16:80) u[80:88) ----
    {
        float h0 = x2[n], dh0 = 1.0f;
        float s[8], ds[8];
#pragma unroll
        for (int k = 0; k < 8; ++k) {
            float a = softplusf_(p0[(size_t)k * E + n] + 0.5413f);
            float b = p0[(size_t)(8 + k) * E + n];
            // u is (8,1): softmax over a singleton == 1
            float C  = fmaf(a, h0, b);
            float sg = sigmoidf_(C);
            s[k]  = sg;
            ds[k] = sg * (1.0f - sg) * a * dh0;
        }
#pragma unroll
        for (int i = 0; i < 8; ++i) {
            float wr[8], wmax = -INFINITY;
#pragma unroll
            for (int j = 0; j < 8; ++j) { wr[j] = p0[(size_t)(16 + i * 8 + j) * E + n]; wmax = fmaxf(wmax, wr[j]); }
            float wsum = 0.0f;
#pragma unroll
            for (int j = 0; j < 8; ++j) { wr[j] = expf(wr[j] - wmax); wsum += wr[j]; }
            float D = 0.0f, dD = 0.0f;
#pragma unroll
            for (int j = 0; j < 8; ++j) { float wij = wr[j] / wsum; D = fmaf(wij, s[j], D); dD = fmaf(wij, ds[j], dD); }
            float q = D + 1e-50f;             // 1e-50 underflows in f32, same as JAX f32
            float t = 1.0f / q - 1.0f;
            h[i]  = -logf(t);
            dh[i] = dD / (q * q * t);
        }
    }
    // ---- block 1: d0=8, d1=8 : rows a[0:8) b[8:16) w[16:80) u[80:144) ----
    {
        float s[8], ds[8];
#pragma unroll
        for (int k = 0; k < 8; ++k) {
            float a = softplusf_(p1[(size_t)k * E + n] + 0.5413f);
            float b = p1[(size_t)(8 + k) * E + n];
            float ur[8], umax = -INFINITY;
#pragma unroll
            for (int j = 0; j < 8; ++j) { ur[j] = p1[(size_t)(80 + k * 8 + j) * E + n]; umax = fmaxf(umax, ur[j]); }
            float usum = 0.0f;
#pragma unroll
            for (int j = 0; j < 8; ++j) { ur[j] = expf(ur[j] - umax); usum += ur[j]; }
            float acc = 0.0f, dacc = 0.0f;
#pragma unroll
            for (int j = 0; j < 8; ++j) { float uk = ur[j] / usum; acc = fmaf(uk, h[j], acc); dacc = fmaf(uk, dh[j], dacc); }
            float C  = fmaf(a, acc, b);
            float dC = a * dacc;
            float sg = sigmoidf_(C);
            s[k]  = sg;
            ds[k] = sg * (1.0f - sg) * dC;
        }
        float hn[8], dhn[8];
#pragma unroll
        for (int i = 0; i < 8; ++i) {
            float wr[8], wmax = -INFINITY;
#pragma unroll
            for (int j = 0; j < 8; ++j) { wr[j] = p1[(size_t)(16 + i * 8 + j) * E + n]; wmax = fmaxf(wmax, wr[j]); }
            float wsum = 0.0f;
#pragma unroll
            for (int j = 0; j < 8; ++j) { wr[j] = expf(wr[j] - wmax); wsum += wr[j]; }
            float D = 0.0f, dD = 0.0f;
#pragma unroll
            for (int j = 0; j < 8; ++j) { float wij = wr[j] / wsum; D = fmaf(wij, s[j], D); dD = fmaf(wij, ds[j], dD); }
            float q = D + 1e-50f;
            float t = 1.0f / q - 1.0f;
            hn[i]  = -logf(t);
            dhn[i] = dD / (q * q * t);
        }
#pragma unroll
        for (int i = 0; i < 8; ++i) { h[i] = hn[i]; dh[i] = dhn[i]; }
    }
    // ---- block 2: d0=8, d1=1 : rows a[0] b[1] w[2] u[3:11) ----
    float y, g;
    {
        float a = softplusf_(p2[(size_t)0 * E + n] + 0.5413f);
        float b = p2[(size_t)1 * E + n];
        float ur[8], umax = -INFINITY;
#pragma unroll
        for (int j = 0; j < 8; ++j) { ur[j] = p2[(size_t)(3 + j) * E + n]; umax = fmaxf(umax, ur[j]); }
        float usum = 0.0f;
#pragma unroll
        for (int j = 0; j < 8; ++j) { ur[j] = expf(ur[j] - umax); usum += ur[j]; }
        float acc = 0.0f, dacc = 0.0f;
#pragma unroll
        for (int j = 0; j < 8; ++j) { float uj = ur[j] / usum; acc = fmaf(uj, h[j], acc); dacc = fmaf(uj, dh[j], dacc); }
        float C  = fmaf(a, acc, b);
        float sg = sigmoidf_(C);
        float ds = sg * (1.0f - sg) * a * dacc;
        // w is (1,1): softmax == 1 -> D = sg
        float q = sg + 1e-50f;
        float t = 1.0f / q - 1.0f;
        y = -logf(t);
        g = ds / (q * q * t);
    }

    bool m = ((n & 1) == (ii & 1));
    float contrib = m ? 0.0f : logf(g);   // log(where(m,1,g))
    if (!m) xf[n] = y;

    __shared__ float red[256];
    red[threadIdx.x] = contrib;
    __syncthreads();
    for (int off = 128; off > 0; off >>= 1) {
        if (threadIdx.x < off) red[threadIdx.x] += red[threadIdx.x + off];
        __syncthreads();
    }
    if (threadIdx.x == 0) ljd_arr[ii * 4 + blockIdx.x] = red[0];
}

// ------------------------------------------------------------------
// finalize: out[0:1024) = xf ; out[1024] = sum of 8 log-jac block sums
// ------------------------------------------------------------------
__global__ void k_final(const float* __restrict__ xf, const float* __restrict__ ljd_arr,
                        float* __restrict__ out) {
    int i = blockIdx.x * blockDim.x + threadIdx.x;
    if (i < E) out[i] = xf[i];
    if (i == E) {
        float s = 0.0f;
        for (int k = 0; k < 8; ++k) s += ljd_arr[k];
        out[E] = s;
    }
}

extern "C" void kernel_launch(void* const* d_in, const int* in_sizes, int n_in,
                              void* d_out, int out_size, void* d_ws, size_t ws_size,
                              hipStream_t stream) {
    const float* x      = (const float*)d_in[0];
    // d_in[1] = maskflow (bool) -- pattern is fixed parity, recomputed in-kernel
    const float* mlp_w0 = (const float*)d_in[2];
    const float* mlp_b0 = (const float*)d_in[3];
    const float* mlp_w1 = (const float*)d_in[4];
    const float* mlp_b1 = (const float*)d_in[5];
    const float* lfw    = (const float*)d_in[6];
    const float* lfb    = (const float*)d_in[7];
    const float* lmw    = (const float*)d_in[8];
    const float* lmb    = (const float*)d_in[9];
    const float* llw    = (const float*)d_in[10];
    const float* llb    = (const float*)d_in[11];

    float* ws    = (float*)d_ws;
    float* xf    = ws;                  // 1024
    float* x1    = ws + 1024;           // 1024
    float* x2    = ws + 2048;           // 1024
    float* hpart = ws + 3072;           // 16*256 = 4096
    float* h     = ws + 7168;           // 512 (h_a, h_b)
    float* p0    = ws + 7680;           // 90112
    float* p1    = p0 + SZ_FIRST;       // 147456
    float* p2    = p1 + SZ_MID;         // 11264
    float* ljd   = p2 + SZ_LAST;        // 8

    k_init<<<4, 256, 0, stream>>>(x, xf);
    for (int ii = 0; ii < 2; ++ii) {
        k_mask_split<<<4, 256, 0, stream>>>(xf, x1, x2, ii);
        k_mlp0<<<16, 256, 0, stream>>>(x1, mlp_w0, hpart, ii);
        k_mlp1<<<2, 256, 0, stream>>>(hpart, mlp_b0, mlp_w1, mlp_b1, h, ii);
        k_biggemv<<<SZ_FIRST / 128, 256, 0, stream>>>(
            h,       lfw + (size_t)ii * MLPW * SZ_FIRST, lfb + (size_t)ii * SZ_FIRST, p0, SZ_FIRST);
        k_biggemv<<<SZ_MID / 128, 256, 0, stream>>>(
            h,       lmw + (size_t)ii * MLPW * SZ_MID,   lmb + (size_t)ii * SZ_MID,   p1, SZ_MID);
        k_biggemv<<<SZ_LAST / 128, 256, 0, stream>>>(
            h + 256, llw + (size_t)ii * MLPW * SZ_LAST,  llb + (size_t)ii * SZ_LAST,  p2, SZ_LAST);
        k_dsf<<<4, 256, 0, stream>>>(p0, p1, p2, x2, xf, ljd, ii);
    }
    k_final<<<5, 256, 0, stream>>>(xf, ljd, (float*)d_out);
}